// ProtoLayer_26843545600122
// MI455X (gfx1250) — compile-verified
//
#include <hip/hip_runtime.h>

typedef __attribute__((ext_vector_type(2))) float v2f;
typedef __attribute__((ext_vector_type(8))) float v8f;

#define T_N        4
#define WAY        16
#define SHOT       8
#define QUERY      256
#define C_DIM      256
#define WQ         (WAY * QUERY)          /* 4096 query rows per t */
#define ROWS_PER_BLOCK 256                /* 8 waves x 2 tiles x 16 rows */
#define LDS_STRIDE 260                    /* pad 256 -> 260 to kill DS bank conflicts */

__global__ __launch_bounds__(256) void proto_dist_kernel(
    const float* __restrict__ query,      // (T, 4096, 256) f32
    const float* __restrict__ support,    // (T, 128, 256)  f32
    float* __restrict__ out)              // (T, 4096, 16)  f32
{
    __shared__ float proto[WAY * LDS_STRIDE];
    __shared__ float pnormPartial[WAY][16];
    __shared__ float pnorm[WAY];

    const int t         = blockIdx.x / (WQ / ROWS_PER_BLOCK);
    const int blockRow0 = (blockIdx.x % (WQ / ROWS_PER_BLOCK)) * ROWS_PER_BLOCK;
    const int tid       = threadIdx.x;

    // ---- build prototypes (mean over shots) into LDS ----
    const float* sup_t = support + (size_t)t * (WAY * SHOT) * C_DIM;
    for (int idx = tid; idx < WAY * C_DIM; idx += 256) {
        const int w = idx >> 8;           // / C_DIM
        const int c = idx & (C_DIM - 1);
        float s = 0.f;
        #pragma unroll
        for (int sh = 0; sh < SHOT; ++sh)
            s += sup_t[(w * SHOT + sh) * C_DIM + c];
        proto[w * LDS_STRIDE + c] = s * (1.0f / SHOT);
    }
    __syncthreads();

    // ---- prototype squared norms ----
    {
        const int w   = tid >> 4;         // 0..15
        const int seg = tid & 15;         // 0..15  (16 channels each)
        float p = 0.f;
        #pragma unroll
        for (int c = seg * 16; c < seg * 16 + 16; ++c) {
            const float v = proto[w * LDS_STRIDE + c];
            p += v * v;
        }
        pnormPartial[w][seg] = p;
    }
    __syncthreads();
    if (tid < WAY) {
        float s = 0.f;
        #pragma unroll
        for (int i = 0; i < 16; ++i) s += pnormPartial[tid][i];
        pnorm[tid] = s;
    }
    __syncthreads();

    // ---- WMMA tiles: 16 query rows x 16 ways per wave-tile ----
    const int wave = tid >> 5;            // 0..7 (wave32)
    const int lane = tid & 31;
    const int m    = lane & 15;           // A: query row in tile / B: way index
    const int hi   = lane >> 4;           // K-pair select (K0,K1 vs K2,K3)

    const float pn = pnorm[m];            // ||p_m||^2 (column m of the tile)

    for (int tt = 0; tt < 2; ++tt) {
        const int   row0 = blockRow0 + (wave * 2 + tt) * 16;
        const float* qrow = query + ((size_t)t * WQ + row0 + m) * C_DIM;
        const float* brow = &proto[m * LDS_STRIDE];

        v8f   acc = {};
        float qn  = 0.f;                  // partial ||q_m||^2 (this lane's K slices)

        #pragma unroll 8
        for (int k0 = 0; k0 < C_DIM; k0 += 4) {
            // A 16x4 f32: V0 = {K0 | K2}, V1 = {K1 | K3} across lane halves
            v2f a = *(const v2f*)(qrow + k0 + 2 * hi);
            // B 4x16 f32: rows striped across lanes, same K split
            v2f b = *(const v2f*)(brow + k0 + 2 * hi);
            qn += a.x * a.x + a.y * a.y;
            acc = __builtin_amdgcn_wmma_f32_16x16x4_f32(
                /*neg_a=*/false, a, /*neg_b=*/false, b,
                /*c_mod=*/(short)0, acc, /*reuse_a=*/false, /*reuse_b=*/false);
        }
        // combine the two K-halves of ||q||^2 (wave32 xor-shuffle)
        qn += __shfl_xor(qn, 16, 32);

        // D layout: VGPR r -> row r (lanes 0-15) / row r+8 (lanes 16-31), N = lane&15
        float* orow = out + ((size_t)t * WQ + row0) * WAY;
        #pragma unroll
        for (int r = 0; r < 8; ++r) {
            const int   rowM = r + 8 * hi;
            const float qn_r = __shfl(qn, rowM, 32);   // lane rowM holds full ||q_rowM||^2
            orow[rowM * WAY + m] = 2.0f * acc[r] - qn_r - pn;
        }
    }
}

extern "C" void kernel_launch(void* const* d_in, const int* in_sizes, int n_in,
                              void* d_out, int out_size, void* d_ws, size_t ws_size,
                              hipStream_t stream) {
    (void)in_sizes; (void)n_in; (void)d_ws; (void)ws_size; (void)out_size;
    const float* query   = (const float*)d_in[0];  // (4,4096,256) f32
    const float* support = (const float*)d_in[1];  // (4,128,256)  f32
    // d_in[2] (support_target) is unused by the reference math.
    float* out = (float*)d_out;                    // (4,4096,16)  f32

    const int blocks = T_N * (WQ / ROWS_PER_BLOCK);  // 64
    proto_dist_kernel<<<blocks, 256, 0, stream>>>(query, support, out);
}